// TwoRobots_56160992362976
// MI455X (gfx1250) — compile-verified
//
#include <hip/hip_runtime.h>
#include <cstdint>

// ---------------------------------------------------------------------------
// TwoRobots Euler step, MI455X (gfx1250).
//
// Pure elementwise stream: 80 MB read + 32 MB write = 112 MB per call.
// Roofline @ 23.3 TB/s HBM => ~4.8 us. No GEMM structure (the reference's
// "selection matmuls" are row gathers), so WMMA is inapplicable; the win is
// (a) killing all transcendentals analytically:
//       cos(atan2(dy,dx)) = dx / sqrt(dx^2+dy^2)
//   =>  Fk*cos(theta_p) = KS*dx  (exact, incl. the 0/0 corner -> 0)
//   =>  Fgx = -(KS*dx + CD*vx),  Fgy = -(KS*dy + CD*vy)
// (b) using the CDNA5 async global->LDS DMA path (ASYNCcnt) to stage the 20
//     input rows with zero VGPR staging, and non-temporal B128 stores for the
//     write-once output stream (don't pollute the 192 MB L2).
//
// Each lane stages 20 x 16B (its own float4 column of rows x[0..7], u[0..3],
// w[0..7]) into LDS via global_load_async_to_lds_b128, waits ASYNCcnt==0
// (each lane reads back only bytes its own async ops wrote -> no barrier),
// computes ~30 FMAs, and streams out 8 x 16B.
//
// Tile: 128 lanes * 20 rows * 16 B = 40 KB LDS -> up to 8 blocks/WGP of the
// 320 KB WGP LDS; 128 threads = 4 wave32 per block -> 32 waves/WGP possible.
// ---------------------------------------------------------------------------

#define BLOCK 128   // threads per block == float4 columns per tile
#define NROWS 20    // 8 rows of x, 4 rows of u, 8 rows of w

typedef __attribute__((ext_vector_type(4))) float v4f;

// Async DMA: 16 bytes global -> LDS per lane. GVS addressing:
//   mem_addr = SGPR_base + VGPR_offset(bytes);  LDS byte address comes from the VDST VGPR.
// Tracked by ASYNCcnt; completion enforced with s_wait_asynccnt.
__device__ __forceinline__ void async_b128(uint32_t lds_byte_addr,
                                           const void* sgpr_base,
                                           int vgpr_byte_off) {
  asm volatile("global_load_async_to_lds_b128 %0, %1, %2"
               :: "v"(lds_byte_addr), "v"(vgpr_byte_off), "s"(sgpr_base)
               : "memory");
}

__global__ __launch_bounds__(BLOCK) void
two_robots_step(const float* __restrict__ x, const float* __restrict__ u,
                const float* __restrict__ w, const float* __restrict__ xbar,
                float* __restrict__ out, int B) {
  __shared__ v4f smem[NROWS * BLOCK];   // 40 KB

  constexpr float H  = 0.05f;
  constexpr float KS = 2.0f;   // kspringGround
  constexpr float CD = 2.0f;   // cdampGround
  constexpr float RM1 = 1.0f;  // 1/M1
  constexpr float RM2 = 1.0f;  // 1/M2

  const int ncol4 = B >> 2;                       // B % 4 == 0 (B = 1e6)
  const int tid   = threadIdx.x;
  const int col4  = blockIdx.x * BLOCK + tid;
  // Clamp OOB lanes to a valid column so every async address is in-bounds;
  // only the stores are predicated.
  const int col4c = (col4 < ncol4) ? col4 : (ncol4 - 1);
  const int cb    = col4c * 16;                   // lane's byte offset in a row
  const int rowb  = B * 4;                        // row pitch in bytes

  // Low 32 bits of a flat shared-aperture address == workgroup LDS byte offset.
  const uint32_t lds0 = (uint32_t)(uintptr_t)(&smem[0]);

  // ---- stage tiles: 20 async B128 copies per lane (deep MLP, no VGPR data) --
  #pragma unroll
  for (int r = 0; r < 8; ++r)
    async_b128(lds0 + (uint32_t)(r * BLOCK + tid) * 16u, x, r * rowb + cb);
  #pragma unroll
  for (int r = 0; r < 4; ++r)
    async_b128(lds0 + (uint32_t)((8 + r) * BLOCK + tid) * 16u, u, r * rowb + cb);
  #pragma unroll
  for (int r = 0; r < 8; ++r)
    async_b128(lds0 + (uint32_t)((12 + r) * BLOCK + tid) * 16u, w, r * rowb + cb);

  // Targets: only xbar[0], xbar[1], xbar[4], xbar[5] are used (overlap the DMA).
  const float xb0 = xbar[0], xb1 = xbar[1], xb4 = xbar[4], xb5 = xbar[5];

  asm volatile("s_wait_asynccnt 0x0" ::: "memory");
  // No barrier: each lane consumes exactly the LDS bytes its own copies wrote.

  // ---- gather tile from LDS (ds_load_b128) --------------------------------
  const v4f px1 = smem[ 0 * BLOCK + tid], py1 = smem[ 1 * BLOCK + tid];
  const v4f vx1 = smem[ 2 * BLOCK + tid], vy1 = smem[ 3 * BLOCK + tid];
  const v4f px2 = smem[ 4 * BLOCK + tid], py2 = smem[ 5 * BLOCK + tid];
  const v4f vx2 = smem[ 6 * BLOCK + tid], vy2 = smem[ 7 * BLOCK + tid];
  const v4f u0  = smem[ 8 * BLOCK + tid], u1  = smem[ 9 * BLOCK + tid];
  const v4f u2  = smem[10 * BLOCK + tid], u3  = smem[11 * BLOCK + tid];
  const v4f w0  = smem[12 * BLOCK + tid], w1  = smem[13 * BLOCK + tid];
  const v4f w2  = smem[14 * BLOCK + tid], w3  = smem[15 * BLOCK + tid];
  const v4f w4  = smem[16 * BLOCK + tid], w5  = smem[17 * BLOCK + tid];
  const v4f w6  = smem[18 * BLOCK + tid], w7  = smem[19 * BLOCK + tid];

  // ---- physics (trig-free closed form) ------------------------------------
  const v4f Fgx1 = -(KS * (px1 - xb0) + CD * vx1);
  const v4f Fgy1 = -(KS * (py1 - xb1) + CD * vy1);
  const v4f Fgx2 = -(KS * (px2 - xb4) + CD * vx2);
  const v4f Fgy2 = -(KS * (py2 - xb5) + CD * vy2);

  v4f o[8];
  o[0] = px1 + H * vx1 + w0;                       // position <- velocity
  o[1] = py1 + H * vy1 + w1;
  o[2] = vx1 + H * ((Fgx1 + u0) * RM1) + w2;       // velocity <- forces/M
  o[3] = vy1 + H * ((Fgy1 + u1) * RM1) + w3;
  o[4] = px2 + H * vx2 + w4;
  o[5] = py2 + H * vy2 + w5;
  o[6] = vx2 + H * ((Fgx2 + u2) * RM2) + w6;
  o[7] = vy2 + H * ((Fgy2 + u3) * RM2) + w7;

  // ---- stream out: non-temporal B128 stores (write-once data) -------------
  if (col4 < ncol4) {
    #pragma unroll
    for (int r = 0; r < 8; ++r) {
      v4f* dst = (v4f*)(out + (size_t)r * (size_t)B) + col4;
      __builtin_nontemporal_store(o[r], dst);
    }
  }
}

extern "C" void kernel_launch(void* const* d_in, const int* in_sizes, int n_in,
                              void* d_out, int out_size, void* d_ws, size_t ws_size,
                              hipStream_t stream) {
  const float* x    = (const float*)d_in[0];   // [8, B]
  const float* u    = (const float*)d_in[1];   // [4, B]
  const float* w    = (const float*)d_in[2];   // [8, B]
  const float* xbar = (const float*)d_in[3];   // [8]
  // d_in[4] == t, unused by the dynamics.

  const int B     = in_sizes[0] / 8;           // 1,000,000 (divisible by 4)
  const int ncol4 = B / 4;
  const int grid  = (ncol4 + BLOCK - 1) / BLOCK;

  two_robots_step<<<grid, BLOCK, 0, stream>>>(x, u, w, xbar, (float*)d_out, B);
}